// DiffClusterMISTcc_bias_54477365182868
// MI455X (gfx1250) — compile-verified
//
#include <hip/hip_runtime.h>
#include <cmath>

#define NPTS 6144
#define DIMS 256
#define TILE 16
#define NT   (NPTS / TILE)     /* 384 column tiles / row tiles */
#define NWAVES 8
#define TPW  (NT / NWAVES)     /* 48 column tiles per wave */
#define KSTEPS (DIMS / 4)      /* 64 WMMA k-steps of 4 */

typedef __attribute__((ext_vector_type(2))) float v2f;
typedef __attribute__((ext_vector_type(8))) float v8f;

// ---------------------------------------------------------------------------
// Kernel 1: per-row squared norms + column-0 distances (anchors source).
// ---------------------------------------------------------------------------
__global__ __launch_bounds__(256) void mist_pre(
    const float* __restrict__ x, const float* __restrict__ y,
    float* __restrict__ sqx, float* __restrict__ sqy,
    float* __restrict__ dx0, float* __restrict__ dy0)
{
  int i = blockIdx.x * blockDim.x + threadIdx.x;
  if (i >= NPTS) return;
  const float4* xi = (const float4*)(x + (size_t)i * DIMS);
  const float4* x0 = (const float4*)x;
  const float4* yi = (const float4*)(y + (size_t)i * DIMS);
  const float4* y0 = (const float4*)y;
  float sx = 0.f, s0x = 0.f, dtx = 0.f;
  float sy = 0.f, s0y = 0.f, dty = 0.f;
  for (int d = 0; d < DIMS / 4; ++d) {
    float4 a = xi[d], b = x0[d];
    sx  += a.x*a.x + a.y*a.y + a.z*a.z + a.w*a.w;
    s0x += b.x*b.x + b.y*b.y + b.z*b.z + b.w*b.w;
    dtx += a.x*b.x + a.y*b.y + a.z*b.z + a.w*b.w;
    float4 c = yi[d], e = y0[d];
    sy  += c.x*c.x + c.y*c.y + c.z*c.z + c.w*c.w;
    s0y += e.x*e.x + e.y*e.y + e.z*e.z + e.w*e.w;
    dty += c.x*e.x + c.y*e.y + c.z*e.z + c.w*e.w;
  }
  sqx[i] = sx;
  sqy[i] = sy;
  dx0[i] = sqrtf(fmaxf(sx + s0x - 2.f * dtx, 0.f));
  dy0[i] = sqrtf(fmaxf(sy + s0y - 2.f * dty, 0.f));
}

// Insert candidate (d, j) into ascending top-6 list with (d asc, idx asc)
// ordering — matches jax.lax.top_k(-d) tie-breaking (lower index first).
__device__ __forceinline__ void ins6(float d, int j, float bd[6], int bi[6]) {
  if (d < bd[5] || (d == bd[5] && j < bi[5])) {
    bd[5] = d; bi[5] = j;
#pragma unroll
    for (int u = 5; u > 0; --u) {
      bool sw = (bd[u] < bd[u - 1]) || (bd[u] == bd[u - 1] && bi[u] < bi[u - 1]);
      if (sw) {
        float td = bd[u]; bd[u] = bd[u - 1]; bd[u - 1] = td;
        int   ti = bi[u]; bi[u] = bi[u - 1]; bi[u - 1] = ti;
      }
    }
  }
}

// One 16x16 Gram tile pair via V_WMMA_F32_16X16X4_F32, K = 256 in steps of 4.
// A (rows, from LDS) and B (cols, from global/L2) share the same per-lane
// layout for the symmetric Gram: lane L holds row/col (L%16), K = 2*(L/16)+{0,1}.
__device__ __forceinline__ void gram_tile(
    const float* xa, const float* ya,             // LDS row tiles
    const float* __restrict__ x, const float* __restrict__ y,
    int col, int half, int aoff, v8f& gx, v8f& gy)
{
  const v2f* axp = (const v2f*)(xa + aoff);
  const v2f* ayp = (const v2f*)(ya + aoff);
  const v2f* bxp = (const v2f*)(x + (size_t)col * DIMS + 2 * half);
  const v2f* byp = (const v2f*)(y + (size_t)col * DIMS + 2 * half);
  gx = (v8f){0.f, 0.f, 0.f, 0.f, 0.f, 0.f, 0.f, 0.f};
  gy = (v8f){0.f, 0.f, 0.f, 0.f, 0.f, 0.f, 0.f, 0.f};
#pragma unroll 4
  for (int kk = 0; kk < KSTEPS; ++kk) {
    v2f ax = axp[2 * kk];   // stride 4 floats between k-steps
    v2f bx = bxp[2 * kk];
    v2f ay = ayp[2 * kk];
    v2f by = byp[2 * kk];
    gx = __builtin_amdgcn_wmma_f32_16x16x4_f32(false, ax, false, bx,
                                               (short)0, gx, false, false);
    gy = __builtin_amdgcn_wmma_f32_16x16x4_f32(false, ay, false, by,
                                               (short)0, gy, false, false);
  }
}

// ---------------------------------------------------------------------------
// Kernel 2: per 16-row tile: (pass 1) kNN over d_xy -> kth index -> anchors,
// (pass 2) counts of d_x<=anchor_x, d_y<=anchor_y. Writes one partial per WG.
// ---------------------------------------------------------------------------
__global__ __launch_bounds__(256, 1) void mist_main(
    const float* __restrict__ x, const float* __restrict__ y,
    const float* __restrict__ sqx, const float* __restrict__ sqy,
    const float* __restrict__ dx0, const float* __restrict__ dy0,
    float* __restrict__ partial)
{
  __shared__ float xa[TILE * DIMS];                 // 16 KB
  __shared__ float ya[TILE * DIMS];                 // 16 KB
  __shared__ float tbuf[NWAVES][TILE][TILE + 1];    // per-wave transpose buf
  __shared__ float cd[NWAVES][TILE][6];
  __shared__ int   ci[NWAVES][TILE][6];
  __shared__ float anchX[TILE], anchY[TILE];
  __shared__ int   nxc[TILE], nyc[TILE];
  __shared__ float rowlog[TILE];

  const int tid  = threadIdx.x;
  const int wave = tid >> 5;
  const int lane = tid & 31;
  const int half = lane >> 4;
  const int lm   = lane & 15;
  const int rowbase = blockIdx.x * TILE;

  // Stage the 16-row tile of x and y into LDS (shared A operand).
  for (int t = tid; t < TILE * DIMS / 4; t += 256) {
    ((float4*)xa)[t] = ((const float4*)(x + (size_t)rowbase * DIMS))[t];
    ((float4*)ya)[t] = ((const float4*)(y + (size_t)rowbase * DIMS))[t];
  }
  if (tid < TILE) { nxc[tid] = 0; nyc[tid] = 0; }
  __syncthreads();

  const int aoff = lm * DIMS + 2 * half;

  // Row-side squared norms for this lane's accumulator rows (r + 8*half).
  float rsqx[8], rsqy[8], rsq[8];
#pragma unroll
  for (int r = 0; r < 8; ++r) {
    int row = rowbase + r + 8 * half;
    rsqx[r] = sqx[row];
    rsqy[r] = sqy[row];
    rsq[r]  = rsqx[r] + rsqy[r];
  }

  // ---------------- pass 1: kNN over d_xy ----------------
  float bd[6]; int bi[6];
#pragma unroll
  for (int u = 0; u < 6; ++u) { bd[u] = 3.0e38f; bi[u] = 0x7fffffff; }

  for (int t = 0; t < TPW; ++t) {
    const int colbase = (wave + t * NWAVES) * TILE;
    const int col = colbase + lm;
    v8f gx, gy;
    gram_tile(xa, ya, x, y, col, half, aoff, gx, gy);
    const float csq = sqx[col] + sqy[col];
#pragma unroll
    for (int r = 0; r < 8; ++r) {
      float d2 = rsq[r] + csq - 2.0f * (gx[r] + gy[r]);
      tbuf[wave][r + 8 * half][lm] = sqrtf(fmaxf(d2, 0.0f));
    }
    __syncthreads();
    if (lane < TILE) {                 // lane L scans row L of this tile
      for (int c = 0; c < TILE; ++c)
        ins6(tbuf[wave][lane][c], colbase + c, bd, bi);
    }
    __syncthreads();
  }

  if (lane < TILE) {
#pragma unroll
    for (int u = 0; u < 6; ++u) { cd[wave][lane][u] = bd[u]; ci[wave][lane][u] = bi[u]; }
  }
  __syncthreads();

  // Merge the 8 waves' candidates; kth = 6th-smallest (K+1 incl. self).
  if (tid < TILE) {
    float md[6]; int mi6[6];
#pragma unroll
    for (int u = 0; u < 6; ++u) { md[u] = 3.0e38f; mi6[u] = 0x7fffffff; }
    for (int w = 0; w < NWAVES; ++w)
      for (int u = 0; u < 6; ++u)
        ins6(cd[w][tid][u], ci[w][tid][u], md, mi6);
    int kth = mi6[5];
    anchX[tid] = dx0[kth];   // reference: d_x[kth_idx, 0]
    anchY[tid] = dy0[kth];
  }
  __syncthreads();

  float aXr[8], aYr[8];
#pragma unroll
  for (int r = 0; r < 8; ++r) { aXr[r] = anchX[r + 8 * half]; aYr[r] = anchY[r + 8 * half]; }

  // ---------------- pass 2: counts ----------------
  int cntx[8], cnty[8];
#pragma unroll
  for (int r = 0; r < 8; ++r) { cntx[r] = 0; cnty[r] = 0; }

  for (int t = 0; t < TPW; ++t) {
    const int colbase = (wave + t * NWAVES) * TILE;
    const int col = colbase + lm;
    v8f gx, gy;
    gram_tile(xa, ya, x, y, col, half, aoff, gx, gy);
    const float csx = sqx[col];
    const float csy = sqy[col];
#pragma unroll
    for (int r = 0; r < 8; ++r) {
      float ddx = sqrtf(fmaxf(rsqx[r] + csx - 2.0f * gx[r], 0.0f));
      float ddy = sqrtf(fmaxf(rsqy[r] + csy - 2.0f * gy[r], 0.0f));
      cntx[r] += (aXr[r] - ddx >= 0.0f) ? 1 : 0;   // st_heaviside(anchor - d)
      cnty[r] += (aYr[r] - ddy >= 0.0f) ? 1 : 0;
    }
  }
#pragma unroll
  for (int r = 0; r < 8; ++r) {
    atomicAdd(&nxc[r + 8 * half], cntx[r]);
    atomicAdd(&nyc[r + 8 * half], cnty[r]);
  }
  __syncthreads();
  if (tid < TILE)
    rowlog[tid] = logf((float)nxc[tid] + 1e-7f) + logf((float)nyc[tid] + 1e-7f);
  __syncthreads();
  if (tid == 0) {
    float s = 0.f;
    for (int i = 0; i < TILE; ++i) s += rowlog[i];
    partial[blockIdx.x] = s;
  }
}

// ---------------------------------------------------------------------------
// Kernel 3: deterministic fixed-order reduction + scalar MI.
// ---------------------------------------------------------------------------
__global__ void mist_fin(const float* __restrict__ partial,
                         float* __restrict__ out, float c_log)
{
  if (threadIdx.x == 0 && blockIdx.x == 0) {
    float s = 0.f;
    for (int i = 0; i < NT; ++i) s += partial[i];
    float n_avg_log = s / (float)NPTS;
    const float k_digamma = 1.50611766f;            // digamma(5)
    float mi = (logf((float)NPTS) + c_log + k_digamma - n_avg_log) / 0.69314718f;
    out[0] = fmaxf(mi, 0.0f);
  }
}

extern "C" void kernel_launch(void* const* d_in, const int* in_sizes, int n_in,
                              void* d_out, int out_size, void* d_ws, size_t ws_size,
                              hipStream_t stream) {
  const float* x = (const float*)d_in[0];
  const float* y = (const float*)d_in[1];
  float* ws  = (float*)d_ws;
  float* sqx = ws;
  float* sqy = ws + NPTS;
  float* dx0 = ws + 2 * NPTS;
  float* dy0 = ws + 3 * NPTS;
  float* partial = ws + 4 * NPTS;   // NT floats

  mist_pre<<<NPTS / 256, 256, 0, stream>>>(x, y, sqx, sqy, dx0, dy0);
  mist_main<<<NT, 256, 0, stream>>>(x, y, sqx, sqy, dx0, dy0, partial);

  // c_log = log(cx*cy/cxy) = lgamma(257) - 2*lgamma(129)  (dx=dy=256)
  float c_log = (float)(lgamma(257.0) - 2.0 * lgamma(129.0));
  mist_fin<<<1, 32, 0, stream>>>(partial, (float*)d_out, c_log);
}